// TransformerLayer_63737314673043
// MI455X (gfx1250) — compile-verified
//
#include <hip/hip_runtime.h>
#include <cstddef>

typedef float v2f __attribute__((ext_vector_type(2)));
typedef float v4f __attribute__((ext_vector_type(4)));
typedef float v8f __attribute__((ext_vector_type(8)));

#define WMMA_F32(a, b, c) \
  __builtin_amdgcn_wmma_f32_16x16x4_f32(false, (a), false, (b), (short)0, (c), false, false)

// ---- CDNA5 async global->LDS path (ASYNCcnt), guarded so either toolchain compiles.
#if __has_builtin(__builtin_amdgcn_global_load_async_to_lds_b128) && \
    __has_builtin(__builtin_amdgcn_s_wait_asynccnt)
#define HAVE_ASYNC_LDS 1
typedef int v4i_ __attribute__((vector_size(16)));          // GNU vector: matches builtin's V4i
typedef __attribute__((address_space(1))) v4i_ v4i_glob;    // prints as "__device__" in diags
typedef __attribute__((address_space(3))) v4i_ v4i_lds;
__device__ __forceinline__ void async_cp16(const float* g, float* l) {
  // generic->AS1: same 64-bit address; generic->AS3: low 32 bits are the LDS offset.
  __builtin_amdgcn_global_load_async_to_lds_b128(
      (v4i_glob*)(unsigned long long)g,
      (v4i_lds*)(unsigned int)(unsigned long long)l, 0, 0);
}
#define ASYNC_WAIT(n) __builtin_amdgcn_s_wait_asynccnt(n)
#else
#define HAVE_ASYNC_LDS 0
#endif

__device__ __forceinline__ float dot4(v4f a, v4f b) {
  return a.x * b.x + a.y * b.y + a.z * b.z + a.w * b.w;
}

// ---------------------------------------------------------------------------
// block_attn_res: per token, softmax over {blocks[0], blocks[1], partial}
// of rms-normalized projections, then weighted sum of the raw vectors.
// ---------------------------------------------------------------------------
__global__ __launch_bounds__(256) void bar_kernel(
    const float* __restrict__ blocks, const float* __restrict__ partial,
    const float* __restrict__ pw, const float* __restrict__ nw,
    float* __restrict__ out, int D, size_t blkStride) {
  const int tok = blockIdx.x;
  const int tid = threadIdx.x;
  const size_t row = (size_t)tok * D;
  const int d0 = tid * 4;

  v4f x0 = *(const v4f*)(blocks + row + d0);
  v4f x1 = *(const v4f*)(blocks + blkStride + row + d0);
  v4f x2 = *(const v4f*)(partial + row + d0);
  v4f w = *(const v4f*)(pw + d0);
  v4f g = *(const v4f*)(nw + d0);
  v4f wg = w * g;

  float vals[6];
  vals[0] = dot4(x0, x0);
  vals[1] = dot4(x1, x1);
  vals[2] = dot4(x2, x2);
  vals[3] = dot4(wg, x0);
  vals[4] = dot4(wg, x1);
  vals[5] = dot4(wg, x2);

#pragma unroll
  for (int j = 0; j < 6; ++j)
#pragma unroll
    for (int off = 16; off >= 1; off >>= 1)
      vals[j] += __shfl_xor(vals[j], off, 32);

  __shared__ float red[8][6];
  const int wave = tid >> 5, lane = tid & 31;
  if (lane == 0) {
#pragma unroll
    for (int j = 0; j < 6; ++j) red[wave][j] = vals[j];
  }
  __syncthreads();
  float tot[6] = {0.f, 0.f, 0.f, 0.f, 0.f, 0.f};
#pragma unroll
  for (int w8 = 0; w8 < 8; ++w8)
#pragma unroll
    for (int j = 0; j < 6; ++j) tot[j] += red[w8][j];

  const float invD = 1.0f / (float)D;
  float l0 = tot[3] * rsqrtf(tot[0] * invD + 1e-6f);
  float l1 = tot[4] * rsqrtf(tot[1] * invD + 1e-6f);
  float l2 = tot[5] * rsqrtf(tot[2] * invD + 1e-6f);
  float mx = fmaxf(l0, fmaxf(l1, l2));
  float e0 = __expf(l0 - mx), e1 = __expf(l1 - mx), e2 = __expf(l2 - mx);
  float inv = 1.0f / (e0 + e1 + e2);
  float a0 = e0 * inv, a1 = e1 * inv, a2 = e2 * inv;

  v4f o = x0 * a0 + x1 * a1 + x2 * a2;
  *(v4f*)(out + row + d0) = o;
}

// ---------------------------------------------------------------------------
// In-place RMSNorm: one block per token.
// ---------------------------------------------------------------------------
__global__ __launch_bounds__(256) void rmsnorm_kernel(float* __restrict__ x,
                                                      const float* __restrict__ w,
                                                      int D) {
  const int tok = blockIdx.x;
  const int tid = threadIdx.x;
  const size_t row = (size_t)tok * D;
  const int d0 = tid * 4;
  v4f xv = *(const v4f*)(x + row + d0);
  v4f wv = *(const v4f*)(w + d0);
  float ss = dot4(xv, xv);
#pragma unroll
  for (int off = 16; off >= 1; off >>= 1) ss += __shfl_xor(ss, off, 32);
  __shared__ float red[8];
  const int wave = tid >> 5, lane = tid & 31;
  if (lane == 0) red[wave] = ss;
  __syncthreads();
  float tot = 0.f;
#pragma unroll
  for (int w8 = 0; w8 < 8; ++w8) tot += red[w8];
  float rinv = rsqrtf(tot / (float)D + 1e-6f);
  *(v4f*)(x + row + d0) = wv * xv * rinv;
}

// ---------------------------------------------------------------------------
// fp32 WMMA GEMM:  out[m,n] = sum_k A[m,k] * W[n,k]  (+ bias / relu / res)
// 64x64 tile per workgroup, 8 waves (each 16x32 of C), BK=16.
// Async global->LDS double buffering when available, else register prefetch.
// ---------------------------------------------------------------------------
enum { EPI_BIAS = 0, EPI_QKV = 1, EPI_RELU = 2, EPI_RES = 3 };

__device__ __forceinline__ void mm_tile(const float (*Asb)[20], const float (*Wsb)[20],
                                        int wm, int wn, int hf, int l16,
                                        v8f& c0, v8f& c1) {
#pragma unroll
  for (int kk = 0; kk < 16; kk += 4) {
    const int cc = kk + 2 * hf;  // f32 frag: lanes 0-15 K={kk,kk+1}, 16-31 K={kk+2,kk+3}
    v2f a = *(const v2f*)(&Asb[wm + l16][cc]);
    v2f b0 = *(const v2f*)(&Wsb[wn + l16][cc]);
    v2f b1 = *(const v2f*)(&Wsb[wn + 16 + l16][cc]);
    c0 = WMMA_F32(a, b0, c0);
    c1 = WMMA_F32(a, b1, c1);
  }
}

template <int EPI>
__global__ __launch_bounds__(256) void gemm_kernel(
    const float* __restrict__ A, const float* __restrict__ W,
    const float* __restrict__ bias, const float* __restrict__ res,
    float* __restrict__ out, int M, int N, int K, int T, int HD) {
  const int tid = threadIdx.x;
  const int wave = tid >> 5, lane = tid & 31;
  const int hf = lane >> 4, l16 = lane & 15;
  const int m0 = blockIdx.x * 64, n0 = blockIdx.y * 64;
  const int wm = (wave & 3) * 16;
  const int wn = (wave >> 2) * 32;
  const int lr = tid >> 2;        // cooperative-load row 0..63
  const int lc = (tid & 3) * 4;   // cooperative-load col 0,4,8,12

  const float* gA = A + (size_t)(m0 + lr) * K + lc;
  const float* gW = W + (size_t)(n0 + lr) * K + lc;

  v8f c0 = {0.f, 0.f, 0.f, 0.f, 0.f, 0.f, 0.f, 0.f};
  v8f c1 = c0;

#if HAVE_ASYNC_LDS
  __shared__ float As[2][64][20];  // stride 20: conflict-free, rows stay 16B aligned
  __shared__ float Ws[2][64][20];
  async_cp16(gA, &As[0][lr][lc]);  // prologue: tile 0 -> buf 0  (2 async ops/thread)
  async_cp16(gW, &Ws[0][lr][lc]);
  int buf = 0;
  for (int k0 = 0; k0 < K; k0 += 16, buf ^= 1) {
    if (k0 + 16 < K) {
      async_cp16(gA + k0 + 16, &As[buf ^ 1][lr][lc]);
      async_cp16(gW + k0 + 16, &Ws[buf ^ 1][lr][lc]);
      ASYNC_WAIT(2);  // newest group may remain in flight; older tile has landed
    } else {
      ASYNC_WAIT(0);
    }
    __syncthreads();
    mm_tile(As[buf], Ws[buf], wm, wn, hf, l16, c0, c1);
    __syncthreads();  // buffer reused by the next iteration's issue
  }
#else
  __shared__ float As[64][20];
  __shared__ float Ws[64][20];
  v4f av = *(const v4f*)(gA);
  v4f wv = *(const v4f*)(gW);
  for (int k0 = 0; k0 < K; k0 += 16) {
    __syncthreads();
    *(v4f*)(&As[lr][lc]) = av;
    *(v4f*)(&Ws[lr][lc]) = wv;
    __syncthreads();
    if (k0 + 16 < K) {  // prefetch next tile: global latency hides under WMMA
      av = *(const v4f*)(gA + k0 + 16);
      wv = *(const v4f*)(gW + k0 + 16);
    }
    mm_tile(As, Ws, wm, wn, hf, l16, c0, c1);
  }
#endif

#pragma unroll
  for (int r = 0; r < 8; ++r) {
    const int gm = m0 + wm + hf * 8 + r;  // C layout: lanes 0-15 rows 0-7, 16-31 rows 8-15
#pragma unroll
    for (int t = 0; t < 2; ++t) {
      const int gn = n0 + wn + t * 16 + l16;
      float v = (t == 0 ? c0[r] : c1[r]) + bias[gn];
      if (EPI == EPI_RELU) v = fmaxf(v, 0.0f);
      if (EPI == EPI_RES) v += res[(size_t)gm * N + gn];
      size_t oidx;
      if (EPI == EPI_QKV) {  // write head-split [B,H,T,hd]
        const int bb = gm / T, tt = gm - bb * T;
        const int hh = gn / HD, dd = gn - hh * HD;
        const int H = N / HD;
        oidx = (((size_t)bb * H + hh) * T + tt) * HD + dd;
      } else {
        oidx = (size_t)gm * N + gn;
      }
      out[oidx] = v;
    }
  }
}

// ---------------------------------------------------------------------------
// Causal flash attention, fp32 WMMA. Q/K/V in [B,H,T,hd] (hd=256).
// 8 waves/WG; each wave owns 16 queries (Q frags resident in VGPRs),
// K/V double-buffered in LDS (async when available), online softmax.
// ---------------------------------------------------------------------------
__global__ __launch_bounds__(256) void flash_kernel(
    const float* __restrict__ Q, const float* __restrict__ Km,
    const float* __restrict__ Vm, float* __restrict__ ctx, int T, int HD, int H,
    float scale) {
#if HAVE_ASYNC_LDS
  __shared__ float Ks[2][16][260];
  __shared__ float Vs[2][16][260];
#else
  __shared__ float Ks[1][16][260];
  __shared__ float Vs[1][16][260];
#endif
  __shared__ float Ps[8][16][18];  // per-wave P tile (C-layout -> A-layout)
  const int tid = threadIdx.x, wave = tid >> 5, lane = tid & 31;
  const int hf = lane >> 4, l16 = lane & 15;
  const int bh = blockIdx.y;
  const int b = bh / H, h = bh - b * H;
  const int q0wg = blockIdx.x * 128;
  const int q0 = q0wg + wave * 16;
  const size_t base = (size_t)bh * T * HD;

  v2f qf[64];
#pragma unroll
  for (int i = 0; i < 64; ++i)
    qf[i] = *(const v2f*)(Q + base + (size_t)(q0 + l16) * HD + 4 * i + 2 * hf);

  v8f zero8 = {0.f, 0.f, 0.f, 0.f, 0.f, 0.f, 0.f, 0.f};
  v8f o[16];
#pragma unroll
  for (int t = 0; t < 16; ++t) o[t] = zero8;
  float mr[8], lsum[8];
#pragma unroll
  for (int r = 0; r < 8; ++r) {
    mr[r] = -1e30f;
    lsum[r] = 0.f;
  }

  const int lrow = tid >> 4;         // coop load: row 0..15
  const int lcol = (tid & 15) * 16;  // coop load: 16-float chunk
  const int kend = q0wg + 128;
  const float* gK = Km + base + (size_t)lrow * HD + lcol;
  const float* gV = Vm + base + (size_t)lrow * HD + lcol;

#if HAVE_ASYNC_LDS
#pragma unroll
  for (int j = 0; j < 4; ++j) {  // prologue: tile 0 -> buf 0 (8 async ops/thread)
    async_cp16(gK + 4 * j, &Ks[0][lrow][lcol + 4 * j]);
    async_cp16(gV + 4 * j, &Vs[0][lrow][lcol + 4 * j]);
  }
#else
  v4f kreg[4], vreg[4];
#pragma unroll
  for (int j = 0; j < 4; ++j) {
    kreg[j] = *(const v4f*)(gK + 4 * j);
    vreg[j] = *(const v4f*)(gV + 4 * j);
  }
#endif

  int buf = 0;
  for (int k0 = 0; k0 < kend; k0 += 16) {
#if HAVE_ASYNC_LDS
    if (k0 + 16 < kend) {
      const size_t adv = (size_t)(k0 + 16) * HD;
#pragma unroll
      for (int j = 0; j < 4; ++j) {
        async_cp16(gK + adv + 4 * j, &Ks[buf ^ 1][lrow][lcol + 4 * j]);
        async_cp16(gV + adv + 4 * j, &Vs[buf ^ 1][lrow][lcol + 4 * j]);
      }
      ASYNC_WAIT(8);
    } else {
      ASYNC_WAIT(0);
    }
    __syncthreads();
#else
    __syncthreads();
#pragma unroll
    for (int j = 0; j < 4; ++j) {
      *(v4f*)(&Ks[0][lrow][lcol + 4 * j]) = kreg[j];
      *(v4f*)(&Vs[0][lrow][lcol + 4 * j]) = vreg[j];
    }
    __syncthreads();
    if (k0 + 16 < kend) {  // prefetch next tile while this one computes
      const size_t adv = (size_t)(k0 + 16) * HD;
#pragma unroll
      for (int j = 0; j < 4; ++j) {
        kreg[j] = *(const v4f*)(gK + adv + 4 * j);
        vreg[j] = *(const v4f*)(gV + adv + 4 * j);
      }
    }
#endif

    if (k0 <= q0 + 15) {  // unmasked or diagonal block for this wave
      const float(*Kst)[260] = Ks[buf];
      const float(*Vst)[260] = Vs[buf];

      // S = Q . K^T  (B[d, key] = K[key][d] -> contiguous row reads)
      v8f s = zero8;
#pragma unroll
      for (int i = 0; i < 64; ++i) {
        v2f bk = *(const v2f*)(&Kst[l16][4 * i + 2 * hf]);
        s = WMMA_F32(qf[i], bk, s);
      }

#pragma unroll
      for (int r = 0; r < 8; ++r) {
        float sv = s[r] * scale;
        if (k0 + 15 > q0) {
          const int qi = q0 + hf * 8 + r;
          const int ki = k0 + l16;
          if (ki > qi) sv = -1e30f;
        }
        s[r] = sv;
      }

      float al[8];
#pragma unroll
      for (int r = 0; r < 8; ++r) {
        float rm = s[r];
#pragma unroll
        for (int off = 8; off >= 1; off >>= 1) rm = fmaxf(rm, __shfl_xor(rm, off, 16));
        const float mn = fmaxf(mr[r], rm);
        const float p = __expf(s[r] - mn);
        float rs = p;
#pragma unroll
        for (int off = 8; off >= 1; off >>= 1) rs += __shfl_xor(rs, off, 16);
        al[r] = __expf(mr[r] - mn);
        lsum[r] = lsum[r] * al[r] + rs;
        mr[r] = mn;
        Ps[wave][hf * 8 + r][l16] = p;
      }
#pragma unroll
      for (int t = 0; t < 16; ++t)
#pragma unroll
        for (int r = 0; r < 8; ++r) o[t][r] *= al[r];

      // O += P . V
#pragma unroll
      for (int kk = 0; kk < 16; kk += 4) {
        v2f pa = *(const v2f*)(&Ps[wave][l16][kk + 2 * hf]);
#pragma unroll
        for (int t = 0; t < 16; ++t) {
          v2f bv;
          bv.x = Vst[kk + 2 * hf][t * 16 + l16];
          bv.y = Vst[kk + 2 * hf + 1][t * 16 + l16];
          o[t] = WMMA_F32(pa, bv, o[t]);
        }
      }
    }
#if HAVE_ASYNC_LDS
    __syncthreads();  // all waves done with buf before it is re-targeted
    buf ^= 1;
#endif
  }

  const int C = H * HD;
#pragma unroll
  for (int r = 0; r < 8; ++r) {
    const float inv = 1.0f / lsum[r];
    const int q = q0 + hf * 8 + r;
#pragma unroll
    for (int t = 0; t < 16; ++t)
      ctx[((size_t)b * T + q) * C + h * HD + t * 16 + l16] = o[t][r] * inv;
  }
}

// ---------------------------------------------------------------------------
extern "C" void kernel_launch(void* const* d_in, const int* in_sizes, int n_in,
                              void* d_out, int out_size, void* d_ws, size_t ws_size,
                              hipStream_t stream) {
  (void)in_sizes; (void)n_in; (void)out_size; (void)ws_size;
  constexpr int Bc = 4, T = 2048, D = 1024, H = 4, HD = 256;
  constexpr int M = Bc * T;
  constexpr int FF = 4 * D;
  constexpr size_t SEG = (size_t)M * D;

  const float* blocks = (const float*)d_in[0];
  const float* hidden = (const float*)d_in[1];
  const float* wq = (const float*)d_in[2];
  const float* bq = (const float*)d_in[3];
  const float* wk = (const float*)d_in[4];
  const float* bk = (const float*)d_in[5];
  const float* wv = (const float*)d_in[6];
  const float* bv = (const float*)d_in[7];
  const float* wo = (const float*)d_in[8];
  const float* bo = (const float*)d_in[9];
  const float* attn_norm_w = (const float*)d_in[10];
  const float* attn_res_pw = (const float*)d_in[11];
  const float* attn_res_nw = (const float*)d_in[12];
  const float* w1 = (const float*)d_in[13];
  const float* b1 = (const float*)d_in[14];
  const float* w2 = (const float*)d_in[15];
  const float* b2 = (const float*)d_in[16];
  const float* mlp_norm_w = (const float*)d_in[17];
  const float* mlp_res_pw = (const float*)d_in[18];
  const float* mlp_res_nw = (const float*)d_in[19];

  float* ws = (float*)d_ws;
  float* xn = ws;            // h1 -> xn -> attention context
  float* Qb = ws + SEG;      // Q -> later h2/x2
  float* Kb = ws + 2 * SEG;  // K -> later MLP mid chunk
  float* Vb = ws + 3 * SEG;  // V
  float* p2 = ws + 4 * SEG;  // partial after attention

  float* outBlocks = (float*)d_out;
  float* outPartial = (float*)d_out + (size_t)2 * M * D;

  (void)hipMemcpyAsync(outBlocks, blocks, (size_t)2 * M * D * sizeof(float),
                       hipMemcpyDeviceToDevice, stream);

  bar_kernel<<<M, 256, 0, stream>>>(blocks, hidden, attn_res_pw, attn_res_nw,
                                    xn, D, (size_t)M * D);
  rmsnorm_kernel<<<M, 256, 0, stream>>>(xn, attn_norm_w, D);

  dim3 gq(M / 64, D / 64);
  gemm_kernel<EPI_QKV><<<gq, 256, 0, stream>>>(xn, wq, bq, nullptr, Qb, M, D, D, T, HD);
  gemm_kernel<EPI_QKV><<<gq, 256, 0, stream>>>(xn, wk, bk, nullptr, Kb, M, D, D, T, HD);
  gemm_kernel<EPI_QKV><<<gq, 256, 0, stream>>>(xn, wv, bv, nullptr, Vb, M, D, D, T, HD);

  flash_kernel<<<dim3(T / 128, Bc * H), 256, 0, stream>>>(Qb, Kb, Vb, xn, T, HD, H,
                                                          0.0625f);

  gemm_kernel<EPI_RES><<<gq, 256, 0, stream>>>(xn, wo, bo, hidden, p2, M, D, D, T, HD);

  bar_kernel<<<M, 256, 0, stream>>>(blocks, p2, mlp_res_pw, mlp_res_nw, Qb, D,
                                    (size_t)M * D);
  rmsnorm_kernel<<<M, 256, 0, stream>>>(Qb, mlp_norm_w, D);

  constexpr int MC = 2048;
  for (int c = 0; c < 4; ++c) {
    const float* a = Qb + (size_t)c * MC * D;
    gemm_kernel<EPI_RELU><<<dim3(MC / 64, FF / 64), 256, 0, stream>>>(
        a, w1, b1, nullptr, Kb, MC, FF, D, T, HD);
    gemm_kernel<EPI_RES><<<dim3(MC / 64, D / 64), 256, 0, stream>>>(
        Kb, w2, b2, p2 + (size_t)c * MC * D, outPartial + (size_t)c * MC * D,
        MC, D, FF, T, HD);
  }
}